// torchLSTM_cell_21620865368070
// MI455X (gfx1250) — compile-verified
//
#include <hip/hip_runtime.h>

// Problem constants (match reference)
#define NT    365
#define NGRID 2000
#define NX    10
#define HS    256           // hidden size H
#define G4    (4 * HS)      // 1024 gate columns

typedef float v2f __attribute__((ext_vector_type(2)));
typedef float v8f __attribute__((ext_vector_type(8)));

__device__ __forceinline__ float fast_sigmoid(float x) {
  return 1.0f / (1.0f + __expf(-x));
}
__device__ __forceinline__ float fast_tanh(float x) {
  // tanh(x) = 1 - 2/(1+e^{2x}); exact limits at +/-inf, no libm call
  return 1.0f - 2.0f / (1.0f + __expf(2.0f * x));
}

// ---------------------------------------------------------------------------
// Init: zero h0 and c state, preload output with bo (atomics accumulate onto
// it), and precompute the step-invariant combined gate bias bih+bhh.
// ---------------------------------------------------------------------------
__global__ void init_kernel(float* __restrict__ h0, float* __restrict__ c,
                            float* __restrict__ out, const float* __restrict__ bo,
                            float* __restrict__ bsum, const float* __restrict__ bih,
                            const float* __restrict__ bhh) {
  int i = blockIdx.x * blockDim.x + threadIdx.x;
  if (i < NGRID * HS) { h0[i] = 0.0f; c[i] = 0.0f; }
  if (i < NT * NGRID) out[i] = bo[0];
  if (i < G4) bsum[i] = bih[i] + bhh[i];
}

// ---------------------------------------------------------------------------
// Input projection for one timestep: x0[g,h] = relu(sum_k x[g,k]*Wi[h,k] + bi[h])
// One block per grid row g, one thread per hidden unit h (K=10 only).
// ---------------------------------------------------------------------------
__global__ __launch_bounds__(HS)
void input_kernel(const float* __restrict__ xt,  // [NGRID, NX]
                  const float* __restrict__ Wi,  // [HS, NX]
                  const float* __restrict__ bi,  // [HS]
                  float* __restrict__ x0) {      // [NGRID, HS]
  __shared__ float WiL[HS * NX];
  __shared__ float xr[NX];
  int tid = threadIdx.x;
  int g = blockIdx.x;
  for (int i = tid; i < HS * NX; i += blockDim.x) WiL[i] = Wi[i];
  if (tid < NX) xr[tid] = xt[(size_t)g * NX + tid];
  __syncthreads();
  float s = bi[tid];
#pragma unroll
  for (int k = 0; k < NX; ++k) s += xr[k] * WiL[tid * NX + k];
  x0[(size_t)g * HS + tid] = fmaxf(s, 0.0f);
}

// ---------------------------------------------------------------------------
// One LSTM step. Grid: (NGRID/16, HS/16). Block: 1 wave (32 threads).
// Each wave computes a 16-row x 16-col tile of all four gates via
// V_WMMA_F32_16X16X4_F32, then does the fused elementwise update + output dot.
// ---------------------------------------------------------------------------
#define SPAD 516   // 512 + 4 pad floats: kills LDS bank conflicts on b64 reads

__global__ __launch_bounds__(32)
void lstm_step_kernel(const float* __restrict__ x0,   // [NGRID, HS]
                      const float* __restrict__ hin,  // [NGRID, HS]
                      float* __restrict__ hout,       // [NGRID, HS]
                      float* __restrict__ c,          // [NGRID, HS]
                      const float* __restrict__ Wih,  // [4H, HS]
                      const float* __restrict__ Whh,  // [4H, HS]
                      const float* __restrict__ bsum, // [4H] = bih + bhh
                      const float* __restrict__ Wo,   // [HS]   (NY == 1)
                      float* __restrict__ outt) {     // [NGRID]
  __shared__ float S[16][SPAD];   // [x0 tile | h tile] : 16 rows x 512 K-cols

  const int lane = threadIdx.x;          // 0..31
  const int g0 = blockIdx.x * 16;        // row-tile base
  const int n0 = blockIdx.y * 16;        // per-gate column-tile base

  // ---- Stage activation tile [x0 | h] into LDS (coalesced b32 loads) ----
  for (int i = lane; i < 16 * HS; i += 32) {
    int r = i >> 8;           // HS == 256
    int col = i & (HS - 1);
    size_t gsrc = (size_t)(g0 + r) * HS + col;
    S[r][col]      = x0[gsrc];
    S[r][HS + col] = hin[gsrc];
  }
  __syncthreads();

  // ---- WMMA fragment indexing (f32 16x16x4 layout) ----
  const int nl    = lane & 15;            // A: M row / B: N col
  const int khalf = (lane >> 4) << 1;     // lanes 0-15 -> K+{0,1}; 16-31 -> K+{2,3}

  v8f acc[4] = {};   // i, f, g, o gate accumulators (16x16 f32 each)

  // Per-gate weight row pointers: gates[:, q*H + n0 + nl] uses W[q*H+n0+nl, :]
#pragma unroll
  for (int ph = 0; ph < 2; ++ph) {
    const float* __restrict__ Wb = ph ? Whh : Wih;
    const float* wq0 = Wb + ((size_t)(0 * HS + n0 + nl)) * HS + khalf;
    const float* wq1 = Wb + ((size_t)(1 * HS + n0 + nl)) * HS + khalf;
    const float* wq2 = Wb + ((size_t)(2 * HS + n0 + nl)) * HS + khalf;
    const float* wq3 = Wb + ((size_t)(3 * HS + n0 + nl)) * HS + khalf;
    const int kbase = ph * HS;
#pragma unroll 4
    for (int k = 0; k < HS; k += 4) {
      v2f a = *(const v2f*)&S[nl][kbase + k + khalf];            // shared by 4 gates
      v2f b0 = *(const v2f*)(wq0 + k);
      v2f b1 = *(const v2f*)(wq1 + k);
      v2f b2 = *(const v2f*)(wq2 + k);
      v2f b3 = *(const v2f*)(wq3 + k);
      acc[0] = __builtin_amdgcn_wmma_f32_16x16x4_f32(false, a, false, b0, (short)0, acc[0], false, false);
      acc[1] = __builtin_amdgcn_wmma_f32_16x16x4_f32(false, a, false, b1, (short)0, acc[1], false, false);
      acc[2] = __builtin_amdgcn_wmma_f32_16x16x4_f32(false, a, false, b2, (short)0, acc[2], false, false);
      acc[3] = __builtin_amdgcn_wmma_f32_16x16x4_f32(false, a, false, b3, (short)0, acc[3], false, false);
    }
  }

  // ---- Fused bias + LSTM elementwise + NY=1 output projection ----
  const int jcol = n0 + nl;
  const float bI = bsum[0 * HS + jcol];
  const float bF = bsum[1 * HS + jcol];
  const float bG = bsum[2 * HS + jcol];
  const float bO = bsum[3 * HS + jcol];
  const float wo = Wo[jcol];
  const int rowadd = (lane >> 4) * 8;   // C/D layout: VGPR r -> M = r (+8 for hi lanes)

#pragma unroll
  for (int r = 0; r < 8; ++r) {
    const int g = g0 + r + rowadd;
    const size_t idx = (size_t)g * HS + jcol;
    float si = fast_sigmoid(acc[0][r] + bI);
    float sf = fast_sigmoid(acc[1][r] + bF);
    float tg = fast_tanh(acc[2][r] + bG);
    float so = fast_sigmoid(acc[3][r] + bO);
    float cn = sf * c[idx] + si * tg;
    float hn = so * fast_tanh(cn);
    c[idx] = cn;
    hout[idx] = hn;
    // out[g] += sum over the 16 N-columns of hn * Wo[col]
    float v = hn * wo;
    v += __shfl_xor(v, 1, 32);
    v += __shfl_xor(v, 2, 32);
    v += __shfl_xor(v, 4, 32);
    v += __shfl_xor(v, 8, 32);
    if (nl == 0) atomicAdd(&outt[g], v);
  }
}

// ---------------------------------------------------------------------------
// Host side: graph-capture friendly (only kernel launches on `stream`).
// ws layout (floats): h0 | h1 | c | x0 | bsum  -> ~8.2 MB
// ---------------------------------------------------------------------------
extern "C" void kernel_launch(void* const* d_in, const int* in_sizes, int n_in,
                              void* d_out, int out_size, void* d_ws, size_t ws_size,
                              hipStream_t stream) {
  (void)in_sizes; (void)n_in; (void)out_size; (void)ws_size;
  const float* x   = (const float*)d_in[0];
  const float* Wi  = (const float*)d_in[1];
  const float* bi  = (const float*)d_in[2];
  const float* Wih = (const float*)d_in[3];
  const float* bih = (const float*)d_in[4];
  const float* Whh = (const float*)d_in[5];
  const float* bhh = (const float*)d_in[6];
  const float* Wo  = (const float*)d_in[7];
  const float* bo  = (const float*)d_in[8];
  float* out = (float*)d_out;

  float* ws = (float*)d_ws;
  const size_t SB = (size_t)NGRID * HS;
  float* h0   = ws;
  float* h1   = ws + SB;
  float* cs   = ws + 2 * SB;
  float* x0   = ws + 3 * SB;
  float* bsum = ws + 4 * SB;

  init_kernel<<<(NT * NGRID + 255) / 256, 256, 0, stream>>>(h0, cs, out, bo,
                                                            bsum, bih, bhh);

  for (int t = 0; t < NT; ++t) {
    input_kernel<<<NGRID, HS, 0, stream>>>(x + (size_t)t * NGRID * NX, Wi, bi, x0);
    const float* hin = (t & 1) ? h1 : h0;
    float*       hot = (t & 1) ? h0 : h1;
    lstm_step_kernel<<<dim3(NGRID / 16, HS / 16), 32, 0, stream>>>(
        x0, hin, hot, cs, Wih, Whh, bsum, Wo, out + (size_t)t * NGRID);
  }
}